// RRoIAlign_82016695485111
// MI455X (gfx1250) — compile-verified
//
#include <hip/hip_runtime.h>
#include <hip/hip_bf16.h>
#include <math.h>

// RRoIAlign for MI455X (gfx1250).
// Bandwidth-bound gather op (~1.3 GB traffic, <1 GFLOP -> no matrix ops; WMMA
// would only add waste). Strategy:
//   1) Transpose features NCHW -> NHWC in workspace using the CDNA5 async
//      global->LDS path (GLOBAL_LOAD_ASYNC_TO_LDS_B128, ASYNCcnt,
//      s_wait_asynccnt) + float4 stores.
//   2) Per-(roi,bin) block, thread = channel: every bilinear tap is a fully
//      coalesced 1 KB read across the 8 wave32s of the block.

#define FB 4
#define FC 256
#define FH 256
#define FW 256
#define OUTH 7
#define OUTW 7
#define NBINS (OUTH * OUTW)
#define SCALE 0.25f

#define TILE 32
#define TPAD 36   // +4 floats: 144-byte rows -> every tile[j][4g] is 16B aligned

// ---------------------------------------------------------------------------
// Kernel 1: transpose (B,C,H*W) -> (B,H*W,C), f32, via LDS tiles.
// Tile load: one async b128 global->LDS copy per lane (per-lane LDS dest,
// per-lane global addr, tracked by ASYNCcnt). Tile store: one float4 per lane.
// ---------------------------------------------------------------------------
__global__ void __launch_bounds__(256)
transpose_bchw_to_bhwc(const float* __restrict__ in, float* __restrict__ out) {
    __shared__ float tile[TILE][TPAD];

    const int P  = FH * FW;                  // 65536
    const int b  = blockIdx.z;
    const int p0 = blockIdx.x * TILE;        // spatial tile base
    const int c0 = blockIdx.y * TILE;        // channel tile base
    const int t  = threadIdx.x;              // 0..255
    const int j  = t >> 3;                   // 0..31
    const int g  = (t & 7) << 2;             // 0,4,...,28

    // tile[j][g..g+3] = in[b][c0+j][p0+g .. p0+g+3]  (contiguous 16B in memory)
    {
        unsigned lds_off = (unsigned)(size_t)&tile[j][g];
        const float* gp = in + ((size_t)b * FC + (size_t)(c0 + j)) * (size_t)P
                             + (size_t)(p0 + g);
        asm volatile("global_load_async_to_lds_b128 %0, %1, off"
                     :
                     : "v"(lds_off), "v"(gp)
                     : "memory");
    }
    asm volatile("s_wait_asynccnt 0" ::: "memory");
    __syncthreads();

    // out[b][p0+j][c0+g+i] = in[b][c0+g+i][p0+j]  -> one coalesced float4 store
    float4 v;
    v.x = tile[g + 0][j];
    v.y = tile[g + 1][j];
    v.z = tile[g + 2][j];
    v.w = tile[g + 3][j];
    float4* dst = (float4*)(out + ((size_t)b * P + (size_t)(p0 + j)) * (size_t)FC
                                + (size_t)(c0 + g));
    *dst = v;
}

// ---------------------------------------------------------------------------
// Kernel 2: one block per (roi, bin); 256 threads = 8 wave32s, one channel
// per thread. Strides let the same kernel run on NHWC (fast path) or NCHW
// (fallback when workspace is too small).
// ---------------------------------------------------------------------------
__global__ void __launch_bounds__(256)
rroi_kernel(const float* __restrict__ feat, const float* __restrict__ rois,
            float* __restrict__ out, int N,
            long long strideB, long long strideC,
            long long strideY, long long strideX) {
    const int bin = blockIdx.x % NBINS;
    const int n   = blockIdx.x / NBINS;
    const int c   = threadIdx.x;          // 0..255
    if (n >= N) return;

    // ROI params (uniform across block; compiler scalarizes these loads).
    const float* r = rois + (size_t)n * 6;
    const int   b     = (int)r[0];
    const float cx    = r[1] * SCALE;
    const float cy    = r[2] * SCALE;
    const float rw    = fmaxf(r[3] * SCALE, 0.0f);
    const float rh    = fmaxf(r[4] * SCALE, 0.0f);
    const float theta = r[5];

    const float binH = rh * (1.0f / OUTH);
    const float binW = rw * (1.0f / OUTW);
    const int ph = bin / OUTW;
    const int pw = bin % OUTW;
    const float cosT = cosf(theta);
    const float sinT = sinf(theta);

    const float* basep = feat + (size_t)b * strideB + (size_t)c * strideC;

    float acc = 0.0f;
#pragma unroll
    for (int sy = 0; sy < 2; ++sy) {
#pragma unroll
        for (int sx = 0; sx < 2; ++sx) {
            const float yy = binH * ((float)ph + ((float)sy + 0.5f) * 0.5f) - rh * 0.5f;
            const float xx = binW * ((float)pw + ((float)sx + 0.5f) * 0.5f) - rw * 0.5f;
            const float x = xx * cosT + yy * sinT + cx;
            const float y = yy * cosT - xx * sinT + cy;

            const float valid =
                (y > -1.0f && y < (float)FH && x > -1.0f && x < (float)FW) ? 1.0f : 0.0f;

            const float ycl = fminf(fmaxf(y, 0.0f), (float)(FH - 1));
            const float xcl = fminf(fmaxf(x, 0.0f), (float)(FW - 1));
            const float fy = fminf(floorf(ycl), (float)(FH - 2));
            const float fx = fminf(floorf(xcl), (float)(FW - 2));
            const int y0 = (int)fy;
            const int x0 = (int)fx;
            const float ly = ycl - fy;
            const float lx = xcl - fx;
            const float hy = 1.0f - ly;
            const float hx = 1.0f - lx;

            const float* p00 = basep + (size_t)y0 * strideY + (size_t)x0 * strideX;
            const float f00 = p00[0];
            const float f01 = p00[strideX];
            const float f10 = p00[strideY];
            const float f11 = p00[strideY + strideX];

            acc += valid * (hy * hx * f00 + hy * lx * f01 +
                            ly * hx * f10 + ly * lx * f11);
        }
    }

    // out[n][c][ph][pw], divided by S*S = 4.
    out[((size_t)n * FC + c) * NBINS + bin] = acc * 0.25f;
}

// ---------------------------------------------------------------------------
extern "C" void kernel_launch(void* const* d_in, const int* in_sizes, int n_in,
                              void* d_out, int out_size, void* d_ws, size_t ws_size,
                              hipStream_t stream) {
    const float* feat = (const float*)d_in[0];
    const float* rois = (const float*)d_in[1];
    float* out = (float*)d_out;
    const int N = in_sizes[1] / 6;

    const size_t featBytes = (size_t)FB * FC * FH * FW * sizeof(float);

    if (ws_size >= featBytes) {
        // Fast path: transpose to channels-last, then fully coalesced gathers.
        float* featT = (float*)d_ws;
        dim3 tgrid((FH * FW) / TILE, FC / TILE, FB);   // 2048 x 8 x 4
        transpose_bchw_to_bhwc<<<tgrid, 256, 0, stream>>>(feat, featT);

        // NHWC strides (in elements): B: H*W*C, Y: W*C, X: C, C: 1
        rroi_kernel<<<N * NBINS, 256, 0, stream>>>(
            featT, rois, out, N,
            (long long)FH * FW * FC, 1LL,
            (long long)FW * FC, (long long)FC);
    } else {
        // Fallback: gather directly from NCHW.
        rroi_kernel<<<N * NBINS, 256, 0, stream>>>(
            feat, rois, out, N,
            (long long)FC * FH * FW, (long long)FH * FW,
            (long long)FW, 1LL);
    }
}